// EfficientGlobalPointer_5884105196194
// MI455X (gfx1250) — compile-verified
//
#include <hip/hip_runtime.h>
#include <hip/hip_bf16.h>
#include <math.h>

// Problem constants (match reference)
#define Bn 8
#define Sn 1024
#define Hn 1024
#define HEADS 12
#define HEAD 64
#define XCH 128   // 2*HEAD channels of x
#define BCH 24    // 2*HEADS bias channels

typedef __attribute__((ext_vector_type(2))) float v2f;
typedef __attribute__((ext_vector_type(8))) float v8f;

// ---------------------------------------------------------------------------
// Kernel 1: x = ctx @ W1^T + b1   ([8192 x 1024] x [1024 x 128])
// One wave computes one 16x16 tile via V_WMMA_F32_16X16X4_F32, K-loop of 256.
// A layout (16x4 f32): lane<16 -> A[M=lane][K=0..1], lane>=16 -> A[M][K=2..3]
// B layout (4x16 f32): mirrored; per-lane load = aligned float2 for both.
// ---------------------------------------------------------------------------
__global__ void gemm_x_kernel(const float* __restrict__ ctx,
                              const float* __restrict__ W1,
                              const float* __restrict__ b1,
                              float* __restrict__ x) {
  const int wave = blockIdx.x * (blockDim.x >> 5) + (threadIdx.x >> 5);
  const int lane = threadIdx.x & 31;
  const int mt = wave >> 3;          // 0..511  (8192 rows / 16)
  const int nt = wave & 7;           // 0..7    (128 cols / 16)
  const int lo = lane & 15;
  const int hi = lane >> 4;

  const int row = mt * 16 + lo;      // A row (M)
  const int col = nt * 16 + lo;      // B row in W1 (N)
  const float* aptr = ctx + (size_t)row * Hn + 2 * hi;
  const float* bptr = W1  + (size_t)col * Hn + 2 * hi;

  v8f c = {};
  for (int k = 0; k < Hn; k += 4) {
    v2f a  = *(const v2f*)(aptr + k);
    v2f bb = *(const v2f*)(bptr + k);
    c = __builtin_amdgcn_wmma_f32_16x16x4_f32(false, a, false, bb,
                                              (short)0, c, false, false);
  }

  const float bias = b1[col];
  // C layout: VGPR r, lane -> element (M = r + 8*hi, N = lo)
  float* out = x + (size_t)(mt * 16) * XCH + nt * 16 + lo;
#pragma unroll
  for (int r = 0; r < 8; ++r) {
    const int M = r + 8 * hi;
    out[(size_t)M * XCH] = c[r] + bias;
  }
}

// ---------------------------------------------------------------------------
// Kernel 2: RoPE. One thread per (row, pair i in 0..31).
// q[d] = x[2d], k[d] = x[2d+1];  pair i covers d = 2i, 2i+1.
// ---------------------------------------------------------------------------
__global__ void rope_kernel(const float* __restrict__ x,
                            float* __restrict__ qr,
                            float* __restrict__ kr) {
  const int idx = blockIdx.x * blockDim.x + threadIdx.x; // exact: 8192*32
  const int i   = idx & 31;
  const int row = idx >> 5;
  const int pos = row & (Sn - 1);

  const float* xr = x + (size_t)row * XCH + 4 * i;
  const float q0 = xr[0], k0 = xr[1], q1 = xr[2], k1 = xr[3];

  // inv_freq = 10000^(-i/32) = exp(-ln(10000)*i/32)
  const float inv = expf(-9.210340371976184f * (float)i * (1.0f / 32.0f));
  const float ang = (float)pos * inv;
  float s, cth;
  sincosf(ang, &s, &cth);

  float2 qo, ko;
  qo.x = q0 * cth - q1 * s;
  qo.y = q1 * cth + q0 * s;
  ko.x = k0 * cth - k1 * s;
  ko.y = k1 * cth + k0 * s;
  *(float2*)(qr + (size_t)row * HEAD + 2 * i) = qo;
  *(float2*)(kr + (size_t)row * HEAD + 2 * i) = ko;
}

// ---------------------------------------------------------------------------
// Kernel 3: biasT[row][c] = 0.5 * (x[row] . W2[c] + b2[c]),  c in 0..23
// ---------------------------------------------------------------------------
__global__ void bias_kernel(const float* __restrict__ x,
                            const float* __restrict__ W2,
                            const float* __restrict__ b2,
                            float* __restrict__ biasT) {
  const int idx = blockIdx.x * blockDim.x + threadIdx.x; // exact: 8192*24
  const int c   = idx % BCH;
  const int row = idx / BCH;
  const float4* xr = (const float4*)(x  + (size_t)row * XCH);
  const float4* wr = (const float4*)(W2 + (size_t)c   * XCH);
  float acc = 0.0f;
#pragma unroll 8
  for (int j = 0; j < XCH / 4; ++j) {
    const float4 a = xr[j], w = wr[j];
    acc = fmaf(a.x, w.x, acc);
    acc = fmaf(a.y, w.y, acc);
    acc = fmaf(a.z, w.z, acc);
    acc = fmaf(a.w, w.w, acc);
  }
  biasT[(size_t)row * BCH + c] = 0.5f * (acc + b2[c]);
}

// ---------------------------------------------------------------------------
// Kernel 4: logits. One wave per (b, 16x16 m/n tile).
// qk tile computed ONCE (16 f32 WMMAs over K=64), reused for 12 heads.
// Row-side values (mask, odd-channel bias) gathered per-VGPR via wave32 shfl.
// ---------------------------------------------------------------------------
__global__ void logits_kernel(const float* __restrict__ qr,
                              const float* __restrict__ kr,
                              const float* __restrict__ biasT,
                              const int* __restrict__ am,
                              float* __restrict__ out) {
  const int wave = blockIdx.x * (blockDim.x >> 5) + (threadIdx.x >> 5);
  const int lane = threadIdx.x & 31;
  int t = wave;
  const int nt = t & 63; t >>= 6;
  const int mt = t & 63; t >>= 6;
  const int b  = t;                  // 0..7
  const int lo = lane & 15;
  const int hi = lane >> 4;

  const int mrow = mt * 16 + lo;     // row index this lane represents
  const int ncol = nt * 16 + lo;     // col index (N = lo for all C VGPRs)

  const float* qptr = qr + ((size_t)(b * Sn + mrow)) * HEAD + 2 * hi;
  const float* kptr = kr + ((size_t)(b * Sn + ncol)) * HEAD + 2 * hi;

  v8f c = {};
#pragma unroll
  for (int s = 0; s < 16; ++s) {
    v2f a  = *(const v2f*)(qptr + 4 * s);
    v2f bb = *(const v2f*)(kptr + 4 * s);
    c = __builtin_amdgcn_wmma_f32_16x16x4_f32(false, a, false, bb,
                                              (short)0, c, false, false);
  }

  // Per-lane bias rows (lane l holds values for row/col mt*16+lo / nt*16+lo)
  const float* bm_row = biasT + ((size_t)(b * Sn + mrow)) * BCH; // odd ch -> row bias
  const float* bn_row = biasT + ((size_t)(b * Sn + ncol)) * BCH; // even ch -> col bias
  float bmv[HEADS], bnv[HEADS];
#pragma unroll
  for (int h = 0; h < HEADS; ++h) {
    bmv[h] = bm_row[2 * h + 1];
    bnv[h] = bn_row[2 * h];
  }

  const float am_n     = (float)am[b * Sn + ncol];
  const float am_m_all = (float)am[b * Sn + mrow];

  // pre[r] = qk/8 - pad*1e12 - causal*1e12  (head-independent)
  float pre[8];
#pragma unroll
  for (int r = 0; r < 8; ++r) {
    const int src = r + 8 * hi;                       // lane holding row r+8*hi
    const float am_m = __shfl(am_m_all, src, 32);
    const int M = mt * 16 + r + 8 * hi;
    float pen = -(1.0f - am_n * am_m) * 1.0e12f;
    if (M > ncol) pen -= 1.0e12f;                     // tril(k=-1) causal mask
    pre[r] = c[r] * 0.125f + pen;
  }

#pragma unroll
  for (int h = 0; h < HEADS; ++h) {
    const float bn = bnv[h];
    const size_t base = ((size_t)(b * HEADS + h)) * Sn * Sn;
#pragma unroll
    for (int r = 0; r < 8; ++r) {
      const int src = r + 8 * hi;
      const float bm = __shfl(bmv[h], src, 32);
      const int M = mt * 16 + r + 8 * hi;
      out[base + (size_t)M * Sn + ncol] = pre[r] + bn + bm;
    }
  }
}

// ---------------------------------------------------------------------------
extern "C" void kernel_launch(void* const* d_in, const int* in_sizes, int n_in,
                              void* d_out, int out_size, void* d_ws, size_t ws_size,
                              hipStream_t stream) {
  const float* ctx = (const float*)d_in[0];   // (B,S,H) f32
  const int*   am  = (const int*)  d_in[1];   // (B,S) i32
  const float* W1  = (const float*)d_in[2];   // (128,1024) f32
  const float* b1  = (const float*)d_in[3];   // (128,) f32
  const float* W2  = (const float*)d_in[4];   // (24,128) f32
  const float* b2  = (const float*)d_in[5];   // (24,) f32
  float* out = (float*)d_out;                 // (8,12,1024,1024) f32

  char* ws = (char*)d_ws;
  float* x  = (float*)(ws);                                 // 4 MB
  float* qr = (float*)(ws + (size_t)4 * 1024 * 1024);       // 2 MB
  float* kr = (float*)(ws + (size_t)6 * 1024 * 1024);       // 2 MB
  float* bT = (float*)(ws + (size_t)8 * 1024 * 1024);       // 768 KB

  // 1) x = ctx @ W1^T + b1 : 512 m-tiles * 8 n-tiles = 4096 waves
  gemm_x_kernel<<<512, 256, 0, stream>>>(ctx, W1, b1, x);
  // 2) RoPE: 8192*32 threads (exact multiple of 256)
  rope_kernel<<<(Bn * Sn * 32) / 256, 256, 0, stream>>>(x, qr, kr);
  // 3) biasT: 8192*24 threads (exact multiple of 256)
  bias_kernel<<<(Bn * Sn * BCH) / 256, 256, 0, stream>>>(x, W2, b2, bT);
  // 4) logits: 8*64*64 = 32768 waves, 8 waves/block
  logits_kernel<<<4096, 256, 0, stream>>>(qr, kr, bT, am, out);
}